// GAT_47124381172061
// MI455X (gfx1250) — compile-verified
//
#include <hip/hip_runtime.h>
#include <math.h>

// ---------------------------------------------------------------------------
// Types for CDNA5 WMMA f32 16x16x4:  A = 16x4 (2 VGPRs), B = 4x16 (2 VGPRs),
// C/D = 16x16 f32 (8 VGPRs).
// ---------------------------------------------------------------------------
typedef float v2f __attribute__((ext_vector_type(2)));
typedef float v8f __attribute__((ext_vector_type(8)));

#define NEG_SLOPE 0.2f

__device__ __forceinline__ float leaky(float x) {
    return x > 0.f ? x : NEG_SLOPE * x;
}

// Monotonic float atomic-max: lowers to native int atomics (no CAS loop).
__device__ __forceinline__ void atomicMaxF32(float* addr, float v) {
    if (v >= 0.f) {
        atomicMax((int*)addr, __float_as_int(v));
    } else {
        atomicMin((unsigned int*)addr, __float_as_uint(v));
    }
}

// ---------------------------------------------------------------------------
// Exact fp32 GEMM via V_WMMA_F32_16X16X4_F32 with A-fragment reuse.
// A: [M,K] row-major; B: [K,Nout] row-major; Out: [M,Nout].
// One wave computes a 16 x (CT*16) strip: CT accumulators share one A
// fragment per k-step (4x fewer A reads from HBM for layer 1, and CT
// independent back-to-back WMMAs per load hide the accumulate latency).
// M%16==0, K%4==0, Nout==CT*16 here.
// ISA lane layout (CDNA5 ISA):
//   A: lane l, reg r -> A[l%16][k + 2*(l/16) + r]
//   B: lane l, reg r -> B[k + 2*(l/16) + r][l%16]
//   C/D: lane l, reg r -> C[r + 8*(l/16)][l%16]
// ---------------------------------------------------------------------------
template <int CT>
__global__ void wmma_gemm_f32(const float* __restrict__ A,
                              const float* __restrict__ B,
                              float* __restrict__ Out,
                              int M, int K, int Nout) {
    const int wave = threadIdx.x >> 5;
    const int lane = threadIdx.x & 31;
    const int row0 = (blockIdx.x * (blockDim.x >> 5) + wave) << 4;
    const int col0 = blockIdx.y * (CT * 16);
    if (row0 >= M) return;              // wave-uniform: EXEC stays all-ones

    const int half = lane >> 4;         // 0 or 1
    const int l16  = lane & 15;
    const float* arow = A + (size_t)(row0 + l16) * K;
    const int col = col0 + l16;

    v8f acc[CT];
    #pragma unroll
    for (int t = 0; t < CT; ++t) acc[t] = (v8f){};

    for (int k = 0; k < K; k += 4) {
        const int ka = k + 2 * half;
        v2f a;
        a.x = arow[ka];
        a.y = arow[ka + 1];
        const float* brow0 = B + (size_t)ka * Nout + col;
        const float* brow1 = B + (size_t)(ka + 1) * Nout + col;
        #pragma unroll
        for (int t = 0; t < CT; ++t) {
            v2f b;
            b.x = brow0[t * 16];
            b.y = brow1[t * 16];
            acc[t] = __builtin_amdgcn_wmma_f32_16x16x4_f32(
                /*neg_a=*/false, a, /*neg_b=*/false, b,
                /*c_mod=*/(short)0, acc[t], /*reuse_a=*/false, /*reuse_b=*/false);
        }
    }
    #pragma unroll
    for (int t = 0; t < CT; ++t) {
        #pragma unroll
        for (int r = 0; r < 8; ++r) {
            Out[(size_t)(row0 + r + 8 * half) * Nout + col + t * 16] = acc[t][r];
        }
    }
}

// ---------------------------------------------------------------------------
// alpha_s[n,h] = dot(h[n,h,:], a_src[h,:]) ; same for alpha_d. One thread per
// (node, head).
// ---------------------------------------------------------------------------
template <int H, int C>
__global__ void alpha_kernel(const float* __restrict__ h,
                             const float* __restrict__ a_src,
                             const float* __restrict__ a_dst,
                             float* __restrict__ as, float* __restrict__ ad,
                             int n) {
    int gid = blockIdx.x * blockDim.x + threadIdx.x;
    if (gid >= n * H) return;
    const int node = gid / H;
    const int hh = gid - node * H;
    const float* hp = h + (size_t)node * (H * C) + hh * C;
    const float* asw = a_src + hh * C;
    const float* adw = a_dst + hh * C;
    float s = 0.f, d = 0.f;
    #pragma unroll
    for (int c = 0; c < C; ++c) {
        s += hp[c] * asw[c];
        d += hp[c] * adw[c];
    }
    as[gid] = s;
    ad[gid] = d;
}

__global__ void fill_kernel(float* __restrict__ p, float v, int n) {
    int gid = blockIdx.x * blockDim.x + threadIdx.x;
    if (gid < n) p[gid] = v;
}

// out[n, j] = bias[j]  (accumulator init; edge scatter adds on top)
__global__ void init_bias_kernel(float* __restrict__ out,
                                 const float* __restrict__ bias,
                                 int n, int HC) {
    int gid = blockIdx.x * blockDim.x + threadIdx.x;
    if (gid >= n * HC) return;
    out[gid] = bias[gid % HC];
}

// Edge index helper: edges laid out [2,E] flat; self-loops appended virtually.
__device__ __forceinline__ void edge_sd(const int* __restrict__ eidx, int E,
                                        int i, int& s, int& d) {
    if (i < E) { s = eidx[i]; d = eidx[E + i]; }
    else       { s = i - E;   d = i - E; }
}

// Pass 1: m[dst,h] = max over edges of leakyrelu(as[src,h]+ad[dst,h])
template <int H>
__global__ void edge_max_kernel(const int* __restrict__ eidx, int E, int n,
                                const float* __restrict__ as,
                                const float* __restrict__ ad,
                                float* __restrict__ m) {
    int gid = blockIdx.x * blockDim.x + threadIdx.x;
    int tot = (E + n) * H;
    if (gid >= tot) return;
    const int i = gid / H;
    const int hh = gid - i * H;
    int s, d;
    edge_sd(eidx, E, i, s, d);
    float e = leaky(as[s * H + hh] + ad[d * H + hh]);
    atomicMaxF32(&m[d * H + hh], e);
}

// Pass 2: denom[dst,h] = sum exp(e - m[dst,h])
template <int H>
__global__ void edge_sum_kernel(const int* __restrict__ eidx, int E, int n,
                                const float* __restrict__ as,
                                const float* __restrict__ ad,
                                const float* __restrict__ m,
                                float* __restrict__ den) {
    int gid = blockIdx.x * blockDim.x + threadIdx.x;
    int tot = (E + n) * H;
    if (gid >= tot) return;
    const int i = gid / H;
    const int hh = gid - i * H;
    int s, d;
    edge_sd(eidx, E, i, s, d);
    float e = leaky(as[s * H + hh] + ad[d * H + hh]);
    float ex = __expf(e - m[d * H + hh]);
    atomicAdd(&den[d * H + hh], ex);
}

// Pass 3: out[dst,h,:] += coeff * h[src,h,:]
template <int H, int C>
__global__ void edge_agg_kernel(const int* __restrict__ eidx, int E, int n,
                                const float* __restrict__ as,
                                const float* __restrict__ ad,
                                const float* __restrict__ m,
                                const float* __restrict__ den,
                                const float* __restrict__ h,
                                float* __restrict__ out) {
    int gid = blockIdx.x * blockDim.x + threadIdx.x;
    int tot = (E + n) * H;
    if (gid >= tot) return;
    const int i = gid / H;
    const int hh = gid - i * H;
    int s, d;
    edge_sd(eidx, E, i, s, d);
    float e = leaky(as[s * H + hh] + ad[d * H + hh]);
    float coeff = __expf(e - m[d * H + hh]) / den[d * H + hh];
    const float* hs = h + (size_t)s * (H * C) + hh * C;
    float* od = out + (size_t)d * (H * C) + hh * C;
    #pragma unroll
    for (int c = 0; c < C; ++c) {
        atomicAdd(&od[c], coeff * hs[c]);
    }
}

// ---------------------------------------------------------------------------
// Host-side orchestration
// ---------------------------------------------------------------------------
static inline int ceil_div(int a, int b) { return (a + b - 1) / b; }

extern "C" void kernel_launch(void* const* d_in, const int* in_sizes, int n_in,
                              void* d_out, int out_size, void* d_ws, size_t ws_size,
                              hipStream_t stream) {
    const float* x    = (const float*)d_in[0];
    const int*   eidx = (const int*)  d_in[1];
    const float* W1   = (const float*)d_in[2];
    const float* aS1  = (const float*)d_in[3];
    const float* aD1  = (const float*)d_in[4];
    const float* b1   = (const float*)d_in[5];
    const float* W2   = (const float*)d_in[6];
    const float* aS2  = (const float*)d_in[7];
    const float* aD2  = (const float*)d_in[8];
    const float* b2   = (const float*)d_in[9];
    float* out = (float*)d_out;

    const int FEATS = 256, HID = 64, H1 = 8, C1 = 8, CLS = 16;
    const int N = in_sizes[0] / FEATS;   // 100000
    const int E = in_sizes[1] / 2;       // 1600000
    const int Etot = E + N;

    // Workspace carve-up (floats)
    float* ws = (float*)d_ws;
    float* h1   = ws;                 ws += (size_t)N * HID;   // [N,8,8]
    float* as1  = ws;                 ws += (size_t)N * H1;
    float* ad1  = ws;                 ws += (size_t)N * H1;
    float* m1   = ws;                 ws += (size_t)N * H1;
    float* den1 = ws;                 ws += (size_t)N * H1;
    float* o1   = ws;                 ws += (size_t)N * HID;   // layer-1 output
    float* h2   = ws;                 ws += (size_t)N * CLS;   // [N,1,16]
    float* as2  = ws;                 ws += (size_t)N;
    float* ad2  = ws;                 ws += (size_t)N;
    float* m2   = ws;                 ws += (size_t)N;
    float* den2 = ws;                 ws += (size_t)N;

    const int TPB = 256;
    const int WPB = 4;                // waves per block in GEMM
    const dim3 gblock(32 * WPB);

    // ---------------- Layer 1 ----------------
    {
        // CT=4: one wave computes a full 16x64 strip, A read once from HBM.
        dim3 grid(ceil_div(N / 16, WPB), 1);
        wmma_gemm_f32<4><<<grid, gblock, 0, stream>>>(x, W1, h1, N, FEATS, HID);
    }
    alpha_kernel<H1, C1><<<ceil_div(N * H1, TPB), TPB, 0, stream>>>(h1, aS1, aD1, as1, ad1, N);
    fill_kernel<<<ceil_div(N * H1, TPB), TPB, 0, stream>>>(m1, -INFINITY, N * H1);
    fill_kernel<<<ceil_div(N * H1, TPB), TPB, 0, stream>>>(den1, 0.f, N * H1);
    init_bias_kernel<<<ceil_div(N * HID, TPB), TPB, 0, stream>>>(o1, b1, N, HID);

    {
        const int tot = Etot * H1;
        edge_max_kernel<H1><<<ceil_div(tot, TPB), TPB, 0, stream>>>(eidx, E, N, as1, ad1, m1);
        edge_sum_kernel<H1><<<ceil_div(tot, TPB), TPB, 0, stream>>>(eidx, E, N, as1, ad1, m1, den1);
        edge_agg_kernel<H1, C1><<<ceil_div(tot, TPB), TPB, 0, stream>>>(eidx, E, N, as1, ad1, m1, den1, h1, o1);
    }

    // ---------------- Layer 2 ----------------
    {
        dim3 grid(ceil_div(N / 16, WPB), 1);
        wmma_gemm_f32<1><<<grid, gblock, 0, stream>>>(o1, W2, h2, N, HID, CLS);
    }
    alpha_kernel<1, CLS><<<ceil_div(N, TPB), TPB, 0, stream>>>(h2, aS2, aD2, as2, ad2, N);
    fill_kernel<<<ceil_div(N, TPB), TPB, 0, stream>>>(m2, -INFINITY, N);
    fill_kernel<<<ceil_div(N, TPB), TPB, 0, stream>>>(den2, 0.f, N);
    init_bias_kernel<<<ceil_div(N * CLS, TPB), TPB, 0, stream>>>(out, b2, N, CLS);

    {
        const int tot = Etot;  // H = 1
        edge_max_kernel<1><<<ceil_div(tot, TPB), TPB, 0, stream>>>(eidx, E, N, as2, ad2, m2);
        edge_sum_kernel<1><<<ceil_div(tot, TPB), TPB, 0, stream>>>(eidx, E, N, as2, ad2, m2, den2);
        edge_agg_kernel<1, CLS><<<ceil_div(tot, TPB), TPB, 0, stream>>>(eidx, E, N, as2, ad2, m2, den2, h2, out);
    }
}